// Graphprogate_63084479644113
// MI455X (gfx1250) — compile-verified
//
#include <hip/hip_runtime.h>

// GNN SpMM scatter for MI455X (gfx1250):
//   out = segment_sum(features[src] * edge_vals, dst) + bias
//
// Memory-bound (AI ~0.24 FLOP/B); ~61 MB working set is L2-resident (192 MB).
//   Kernel 1: bias broadcast via one V_WMMA_F32_16X16X4_F32, branch-free
//             full-tile store fast path.
//   Kernel 2: double-buffered async edge staging (global_load_async_to_lds_b128,
//             ASYNCcnt) overlapped with compute; one edge per wave32;
//             b128 L2 gather; 4x global_atomic_add_f32 (no-return) per lane.

#define D_FEAT 128
#define TILE_E 1024  // edges per staged tile (256 threads x 4 via b128)

typedef float v2f __attribute__((ext_vector_type(2)));
typedef float v8f __attribute__((ext_vector_type(8)));
typedef int   v4i __attribute__((ext_vector_type(4)));

typedef __attribute__((address_space(1))) v4i GV4;  // global int4
typedef __attribute__((address_space(3))) v4i LV4;  // LDS int4

__device__ __forceinline__ void async_copy_b128(const void* g, void* l) {
#if __has_builtin(__builtin_amdgcn_global_load_async_to_lds_b128)
  __builtin_amdgcn_global_load_async_to_lds_b128((GV4*)g, (LV4*)l, 0, 0);
#else
  *(int4*)l = *(const int4*)g;   // fallback: synchronous staging
#endif
}

__device__ __forceinline__ void async_wait_all() {
#if __has_builtin(__builtin_amdgcn_global_load_async_to_lds_b128) && \
    __has_builtin(__builtin_amdgcn_s_wait_asynccnt)
  __builtin_amdgcn_s_wait_asynccnt(0);
#endif
}

// ---------------------------------------------------------------------------
// Kernel 1: out[n][d] = bias[d] for all n.
//   A (16x4): exactly one 1.0 per row; B (4x16): bias in every K row;
//   C = 0  =>  D[m][n] = bias[n] exactly (f32 WMMA).
// Each of the 8 waves owns one 16-col tile and streams it to all row tiles.
// ---------------------------------------------------------------------------
__global__ __launch_bounds__(256) void gnn_bias_init_wmma(
    const float* __restrict__ bias, float* __restrict__ out, int n_nodes) {
  const int lane  = threadIdx.x & 31;
  const int wave  = threadIdx.x >> 5;        // 8 waves -> 8 column tiles
  const int cbase = wave * 16;
  const int n15   = lane & 15;

  v2f a;
  a.x = (lane < 16) ? 1.0f : 0.0f;           // one K-slot = 1 per A row
  a.y = 0.0f;
  const float bv = bias[cbase + n15];
  v2f b;
  b.x = bv;                                   // bias in every K row of B
  b.y = bv;
  v8f c = {};
#if __has_builtin(__builtin_amdgcn_wmma_f32_16x16x4_f32)
  v8f d = __builtin_amdgcn_wmma_f32_16x16x4_f32(
      false, a, false, b, (short)0, c, false, false);
#else
  v8f d;
  #pragma unroll
  for (int v = 0; v < 8; ++v) d[v] = bv;
#endif

  // C/D layout: VGPR v, lanes 0-15 -> row M=v, lanes 16-31 -> row M=v+8.
  const int mbase       = (lane < 16) ? 0 : 8;
  const int n_row_tiles = (n_nodes + 15) >> 4;
  for (int rt = blockIdx.x; rt < n_row_tiles; rt += gridDim.x) {
    const int rbase = rt << 4;
    float* orow = out + (size_t)(rbase + mbase) * D_FEAT + cbase + n15;
    if (rbase + 16 <= n_nodes) {             // full tile: branch-free stores
      #pragma unroll
      for (int v = 0; v < 8; ++v)
        orow[(size_t)v * D_FEAT] = d[v];
    } else {                                  // tail tile: guarded
      #pragma unroll
      for (int v = 0; v < 8; ++v)
        if (rbase + mbase + v < n_nodes) orow[(size_t)v * D_FEAT] = d[v];
    }
  }
}

// ---------------------------------------------------------------------------
// Kernel 2: scatter with double-buffered async edge staging.
// Per iteration: wait for buffer p (staged last iteration), kick off async
// b128 staging of the next tile into buffer p^1, then consume buffer p:
// one edge per wave32, 32 lanes x float4 = full 128-float row, 4 f32
// atomics/lane. End-of-iteration barrier fences LDS buffer reuse.
// ---------------------------------------------------------------------------
__global__ __launch_bounds__(256) void gnn_spmm_scatter(
    const int* __restrict__ srcs, const int* __restrict__ dsts,
    const float* __restrict__ vals, const float* __restrict__ feats,
    float* __restrict__ out, int n_edges) {
  __shared__ int   s_src[2][TILE_E];
  __shared__ int   s_dst[2][TILE_E];
  __shared__ float s_val[2][TILE_E];

  const int lane    = threadIdx.x & 31;
  const int wave    = threadIdx.x >> 5;
  const int t       = threadIdx.x;
  const int n_tiles = (n_edges + TILE_E - 1) / TILE_E;

  auto stage = [&](int tile, int p) {
    const int ebase = tile * TILE_E;
    const int rem   = n_edges - ebase;       // > 0 by construction
    const int c4    = t * 4;                 // this thread's 4-edge chunk
    if (c4 + 4 <= rem) {                     // full chunk: async b128 x3
      async_copy_b128(srcs + ebase + c4, &s_src[p][c4]);
      async_copy_b128(dsts + ebase + c4, &s_dst[p][c4]);
      async_copy_b128(vals + ebase + c4, &s_val[p][c4]);
    } else if (c4 < rem) {                   // ragged tail: scalar staging
      for (int k = c4; k < rem; ++k) {
        s_src[p][k] = srcs[ebase + k];
        s_dst[p][k] = dsts[ebase + k];
        s_val[p][k] = vals[ebase + k];
      }
    }
  };

  int tile = blockIdx.x;
  int p    = 0;
  if (tile < n_tiles) stage(tile, 0);        // prologue: stage first tile

  for (; tile < n_tiles; tile += gridDim.x) {
    async_wait_all();                        // buffer p loads complete
    __syncthreads();                         // whole tile visible to block

    const int nxt = tile + gridDim.x;
    if (nxt < n_tiles) stage(nxt, p ^ 1);    // overlap next tile's staging

    const int ebase = tile * TILE_E;
    const int n_e = (n_edges - ebase < TILE_E) ? (n_edges - ebase) : TILE_E;
    #pragma unroll 2
    for (int i = wave; i < n_e; i += 8) {    // one edge per wave
      const int   src = s_src[p][i];
      const int   dst = s_dst[p][i];
      const float val = s_val[p][i];

      const float4 f = *reinterpret_cast<const float4*>(
          feats + (size_t)src * D_FEAT + (lane << 2));   // b128 gather (L2)

      float* o = out + (size_t)dst * D_FEAT + (lane << 2);
      (void)__hip_atomic_fetch_add(o + 0, f.x * val, __ATOMIC_RELAXED,
                                   __HIP_MEMORY_SCOPE_AGENT);
      (void)__hip_atomic_fetch_add(o + 1, f.y * val, __ATOMIC_RELAXED,
                                   __HIP_MEMORY_SCOPE_AGENT);
      (void)__hip_atomic_fetch_add(o + 2, f.z * val, __ATOMIC_RELAXED,
                                   __HIP_MEMORY_SCOPE_AGENT);
      (void)__hip_atomic_fetch_add(o + 3, f.w * val, __ATOMIC_RELAXED,
                                   __HIP_MEMORY_SCOPE_AGENT);
    }
    __syncthreads();                         // fence before buffers reused
    p ^= 1;
  }
}

// ---------------------------------------------------------------------------
extern "C" void kernel_launch(void* const* d_in, const int* in_sizes, int n_in,
                              void* d_out, int out_size, void* d_ws,
                              size_t ws_size, hipStream_t stream) {
  const int*   edge_index = (const int*)d_in[0];    // [2, E] row-major
  const float* edge_vals  = (const float*)d_in[1];  // [E]
  const float* features   = (const float*)d_in[2];  // [N, 128]
  const float* bias       = (const float*)d_in[3];  // [128]
  float*       out        = (float*)d_out;          // [N, 128]

  const int E = in_sizes[0] / 2;
  const int N = in_sizes[2] / D_FEAT;

  const int* srcs = edge_index;       // row 0
  const int* dsts = edge_index + E;   // row 1

  // 1) out = broadcast(bias)  (WMMA-generated tile, streamed stores)
  int row_tiles = (N + 15) / 16;
  int grid_init = row_tiles < 512 ? row_tiles : 512;
  if (grid_init < 1) grid_init = 1;
  gnn_bias_init_wmma<<<grid_init, 256, 0, stream>>>(bias, out, N);

  // 2) out[dst] += val * features[src]
  //    ~512 blocks keeps enough waves for L2 latency hiding while giving
  //    each block >=2 tiles so the async double-buffer pipeline is active.
  int n_tiles   = (E + TILE_E - 1) / TILE_E;
  int grid_scat = n_tiles < 512 ? n_tiles : 512;
  if (grid_scat < 1) grid_scat = 1;
  gnn_spmm_scatter<<<grid_scat, 256, 0, stream>>>(srcs, dsts, edge_vals,
                                                  features, out, E);
}